// SparsemaxSelector_8959301780120
// MI455X (gfx1250) — compile-verified
//
#include <hip/hip_runtime.h>
#include <cstdint>

// ---------------- configuration ----------------
#define HB        4096          // histogram bins = top 12 bits of monotonic key
#define TPB       256
#define NBLK      2048
#define CAP       4096          // candidate buffer capacity
#define SORT_N    4096
#define FTPB      512           // finalize block
#define TOPK      256

typedef float v2f __attribute__((ext_vector_type(2)));
typedef float v8f __attribute__((ext_vector_type(8)));

// Monotonic bit transform: order of keys == order of floats.
__device__ __forceinline__ unsigned keyOf(float f) {
  unsigned u = __float_as_uint(f);
  return (u & 0x80000000u) ? ~u : (u | 0x80000000u);
}

// gfx1250 async global->LDS copy, GVS mode: addr = SGPR64 + VGPR32 + 0.
// VDST = per-lane LDS byte offset (wave-relative), 16 bytes per lane.
__device__ __forceinline__ void asyncLoadB128(unsigned ldsOff, unsigned long long sbase,
                                              unsigned voffBytes) {
  asm volatile("global_load_async_to_lds_b128 %0, %1, %2"
               :: "v"(ldsOff), "v"(voffBytes), "s"(sbase)
               : "memory");
}

// ---------------- kernel 0: zero workspace ----------------
__global__ void zero_kernel(unsigned* __restrict__ hist, unsigned* __restrict__ counter) {
  unsigned i = blockIdx.x * blockDim.x + threadIdx.x;
  if (i < HB) hist[i] = 0u;
  if (i == 0) *counter = 0u;
}

// ---------------- kernel 1: streaming histogram via async-to-LDS ----------------
__global__ __launch_bounds__(TPB)
void hist_kernel(const float* __restrict__ in, unsigned n, unsigned* __restrict__ ghist) {
  __shared__ unsigned hist[HB];
  __shared__ float stage[2][TPB * 4];          // double-buffered 4KB tiles

  for (unsigned i = threadIdx.x; i < HB; i += TPB) hist[i] = 0u;
  __syncthreads();

  const unsigned tid     = threadIdx.x;
  const unsigned gstride = gridDim.x * TPB;    // float4 granularity
  const unsigned nvec    = n >> 2;
  const unsigned base    = blockIdx.x * TPB + tid;
  const unsigned iters   = nvec / gstride;

  const unsigned lds0 = (unsigned)(uintptr_t)(&stage[0][tid * 4]);
  const unsigned lds1 = (unsigned)(uintptr_t)(&stage[1][tid * 4]);
  const unsigned long long gbase = (unsigned long long)in;

  if (iters) asyncLoadB128(lds0, gbase, base * 16u);

  for (unsigned it = 0; it < iters; ++it) {
    const unsigned cur = it & 1u;
    if (it + 1 < iters) {
      asyncLoadB128(cur ? lds0 : lds1, gbase, ((it + 1) * gstride + base) * 16u);
      asm volatile("s_wait_asynccnt 1" ::: "memory");   // oldest (cur) complete, in-order
    } else {
      asm volatile("s_wait_asynccnt 0" ::: "memory");
    }
    const float* sp = &stage[cur][tid * 4];             // each lane reads its own 16B
#pragma unroll
    for (int c = 0; c < 4; ++c)
      atomicAdd(&hist[keyOf(sp[c]) >> 20], 1u);
  }
  // vector tail (empty for N = 2^25)
  for (unsigned v = iters * gstride + base; v < nvec; v += gstride) {
    float4 f = ((const float4*)in)[v];
    atomicAdd(&hist[keyOf(f.x) >> 20], 1u);
    atomicAdd(&hist[keyOf(f.y) >> 20], 1u);
    atomicAdd(&hist[keyOf(f.z) >> 20], 1u);
    atomicAdd(&hist[keyOf(f.w) >> 20], 1u);
  }
  // scalar tail
  if (blockIdx.x == 0)
    for (unsigned i = (nvec << 2) + tid; i < n; i += TPB)
      atomicAdd(&hist[keyOf(in[i]) >> 20], 1u);

  __syncthreads();
  for (unsigned i = tid; i < HB; i += TPB) {
    unsigned c = hist[i];
    if (c) atomicAdd(&ghist[i], c);
  }
}

// ---------------- kernel 2: find threshold bin (cum-from-top >= TOPK) ----------------
__global__ __launch_bounds__(256)
void select_kernel(const unsigned* __restrict__ ghist, unsigned* __restrict__ thrOut) {
  __shared__ unsigned h[HB];
  __shared__ unsigned cs[256];
  const unsigned t = threadIdx.x;
  for (unsigned i = t; i < HB; i += 256) h[i] = ghist[i];
  __syncthreads();

  unsigned s = 0;
#pragma unroll
  for (int b = 0; b < 16; ++b) s += h[t * 16 + b];
  cs[t] = s;
  __syncthreads();
  // Hillis-Steele inclusive suffix sum over 256 chunk totals
  for (unsigned off = 1; off < 256; off <<= 1) {
    unsigned add = (t + off < 256) ? cs[t + off] : 0u;
    __syncthreads();
    cs[t] += add;
    __syncthreads();
  }
  unsigned sufNext = (t < 255) ? cs[t + 1] : 0u;
  if (cs[t] >= TOPK && sufNext < TOPK) {      // exactly one thread (cs nonincreasing)
    unsigned cum = sufNext;
    unsigned B = t * 16;
    for (int b = 15; b >= 0; --b) {
      cum += h[t * 16 + b];
      if (cum >= TOPK) { B = t * 16 + (unsigned)b; break; }
    }
    *thrOut = B << 20;                         // min key of threshold bin
  }
}

// ---------------- kernel 3: compact candidates (key >= threshold) ----------------
__global__ __launch_bounds__(TPB)
void compact_kernel(const float* __restrict__ in, unsigned n,
                    const unsigned* __restrict__ thrPtr,
                    float* __restrict__ cv, int* __restrict__ ci,
                    unsigned* __restrict__ counter) {
  const unsigned thr = *thrPtr;
  const unsigned nvec = n >> 2;
  const unsigned gstride = gridDim.x * TPB;
  for (unsigned v = blockIdx.x * TPB + threadIdx.x; v < nvec; v += gstride) {
    float4 f = ((const float4*)in)[v];
    const float vals[4] = {f.x, f.y, f.z, f.w};
#pragma unroll
    for (int c = 0; c < 4; ++c) {
      if (keyOf(vals[c]) >= thr) {
        unsigned slot = atomicAdd(counter, 1u);
        if (slot < CAP) { cv[slot] = vals[c]; ci[slot] = (int)(4u * v + c); }
      }
    }
  }
  if (blockIdx.x == 0)
    for (unsigned i = (nvec << 2) + threadIdx.x; i < n; i += TPB) {
      float x = in[i];
      if (keyOf(x) >= thr) {
        unsigned slot = atomicAdd(counter, 1u);
        if (slot < CAP) { cv[slot] = x; ci[slot] = (int)i; }
      }
    }
}

// ---------------- kernel 4: sort + WMMA prefix-scan + emit indices ----------------
__device__ __forceinline__ bool comesFirst(float v1, int i1, float v2, int i2) {
  // descending by value, ascending by index on ties (matches lax.top_k tie-break)
  return (v1 > v2) || (v1 == v2 && i1 <= i2);
}

__global__ __launch_bounds__(FTPB)
void finalize_kernel(const float* __restrict__ cv, const int* __restrict__ ci,
                     const unsigned* __restrict__ counter, int* __restrict__ out) {
  __shared__ float sv[SORT_N];
  __shared__ int   si[SORT_N];
  __shared__ float gmat[16 * 16];
  __shared__ float pArr[16 * 16];
  __shared__ int   flag[2 * TOPK];
  __shared__ int   kzSh;

  const int tid = threadIdx.x;
  unsigned cnt = *counter;
  if (cnt > CAP) cnt = CAP;

  for (int i = tid; i < SORT_N; i += FTPB) {
    if ((unsigned)i < cnt) { sv[i] = cv[i]; si[i] = ci[i]; }
    else                   { sv[i] = -__builtin_inff(); si[i] = 0x7FFFFFFF; }
  }
  __syncthreads();

  // Bitonic sort: descending by value, index tie-break ascending.
  for (int k = 2; k <= SORT_N; k <<= 1) {
    for (int j = k >> 1; j > 0; j >>= 1) {
      for (int i = tid; i < SORT_N; i += FTPB) {
        int ixj = i ^ j;
        if (ixj > i) {
          bool desc = ((i & k) == 0);
          float v1 = sv[i], v2 = sv[ixj];
          int   a1 = si[i], a2 = si[ixj];
          bool ok = comesFirst(v1, a1, v2, a2);
          if (desc ? !ok : ok) { sv[i] = v2; sv[ixj] = v1; si[i] = a2; si[ixj] = a1; }
        }
      }
      __syncthreads();
    }
  }

  // --- WMMA inclusive prefix sum of top 256 sorted values (scan-as-GEMM) ---
  // A[r][c] = sv[16r+c];  P = A*U + (Lstrict*A)*Ones, via v_wmma_f32_16x16x4_f32.
  // Layouts (ISA 7.12.2): element index K = kb + 2*(lane>>4) + t for A(row=lane&15)
  // and B(col=lane&15); C/D: row = j + 8*(lane>>4), col = lane&15.
  const int L = tid, hh = (L >> 4) & 1, m = L & 15;
  v8f g = {0.f, 0.f, 0.f, 0.f, 0.f, 0.f, 0.f, 0.f};
  if (L < 32) {
#pragma unroll
    for (int kb = 0; kb < 16; kb += 4) {
      const int k0 = kb + 2 * hh;
      v2f aL, bA;                                  // Lstrict (A-op), A (B-op)
      aL.x = (k0     < m) ? 1.f : 0.f;
      aL.y = (k0 + 1 < m) ? 1.f : 0.f;
      bA.x = sv[16 * (k0)     + m];
      bA.y = sv[16 * (k0 + 1) + m];
      g = __builtin_amdgcn_wmma_f32_16x16x4_f32(false, aL, false, bA,
                                                (short)0, g, false, false);
    }
#pragma unroll
    for (int j = 0; j < 8; ++j) gmat[(j + 8 * hh) * 16 + m] = g[j];   // C-layout -> LDS
  }
  __syncthreads();                                 // cross-lane transpose via LDS
  if (L < 32) {
    v8f acc = {0.f, 0.f, 0.f, 0.f, 0.f, 0.f, 0.f, 0.f};
#pragma unroll
    for (int kb = 0; kb < 16; kb += 4) {           // (Lstrict*A) * Ones
      const int k0 = kb + 2 * hh;
      v2f aG, b1;
      aG.x = gmat[m * 16 + k0];
      aG.y = gmat[m * 16 + k0 + 1];
      b1.x = 1.f; b1.y = 1.f;
      acc = __builtin_amdgcn_wmma_f32_16x16x4_f32(false, aG, false, b1,
                                                  (short)0, acc, false, false);
    }
#pragma unroll
    for (int kb = 0; kb < 16; kb += 4) {           // + A * U
      const int k0 = kb + 2 * hh;
      v2f aA, bU;
      aA.x = sv[16 * m + k0];
      aA.y = sv[16 * m + k0 + 1];
      bU.x = (k0     <= m) ? 1.f : 0.f;
      bU.y = (k0 + 1 <= m) ? 1.f : 0.f;
      acc = __builtin_amdgcn_wmma_f32_16x16x4_f32(false, aA, false, bU,
                                                  (short)0, acc, false, false);
    }
#pragma unroll
    for (int j = 0; j < 8; ++j) pArr[(j + 8 * hh) * 16 + m] = acc[j]; // p[16r+c]
  }
  if (tid == 0) kzSh = 0;
  __syncthreads();

  // Support size: k_z = #{ i : z_sorted[i] - (cumsum[i]-1)/(i+1) > 0 }
  if (tid < TOPK) {
    float zi = sv[tid], pi = pArr[tid];
    if (zi - (pi - 1.0f) / (float)(tid + 1) > 0.0f) atomicAdd(&kzSh, 1);
  }
  if (tid < 2 * TOPK) flag[tid] = 0;
  __syncthreads();

  int kz = kzSh;
  if (kz < 1) kz = 1;
  if (kz > TOPK) kz = TOPK;
  if ((unsigned)kz > cnt) kz = (int)cnt;

  // Mark small support indices so the zero-prob filler can skip them.
  if (tid < kz) { int ii = si[tid]; if (ii >= 0 && ii < 2 * TOPK) flag[ii] = 1; }
  __syncthreads();

  if (tid == 0) {
    int pos = 0;
    for (int j = 0; j < kz && pos < TOPK; ++j) out[pos++] = si[j];   // support, desc
    int nxt = 0;
    while (pos < TOPK) {                                             // zeros: lowest idx
      if (!flag[nxt]) out[pos++] = nxt;
      ++nxt;
    }
  }
}

// ---------------- launch ----------------
extern "C" void kernel_launch(void* const* d_in, const int* in_sizes, int n_in,
                              void* d_out, int out_size, void* d_ws, size_t ws_size,
                              hipStream_t stream) {
  const float* scores = (const float*)d_in[0];
  const unsigned n = (unsigned)in_sizes[0];

  char* ws = (char*)d_ws;
  unsigned* hist    = (unsigned*)(ws);                 // 16384 B
  unsigned* counter = (unsigned*)(ws + 16384);         // 4 B
  unsigned* thrKey  = (unsigned*)(ws + 16388);         // 4 B
  float*    cv      = (float*)(ws + 16400);            // CAP*4
  int*      ci      = (int*)  (ws + 16400 + 4 * CAP);  // CAP*4
  int*      out     = (int*)d_out;

  zero_kernel    <<<(HB + 255) / 256, 256, 0, stream>>>(hist, counter);
  hist_kernel    <<<NBLK, TPB, 0, stream>>>(scores, n, hist);
  select_kernel  <<<1, 256, 0, stream>>>(hist, thrKey);
  compact_kernel <<<NBLK, TPB, 0, stream>>>(scores, n, thrKey, cv, ci, counter);
  finalize_kernel<<<1, FTPB, 0, stream>>>(cv, ci, counter, out);
}